// image2patch_63840393888313
// MI455X (gfx1250) — compile-verified
//
#include <hip/hip_runtime.h>
#include <stdint.h>

// Geometry (compile-time constants from the reference)
#define IMG          256
#define PSIZE        6
#define NPOS         126                 // CUR[q] == 2*q, q in [0,126)
#define TILE_FLOATS  (PSIZE * IMG)       // 6 contiguous image rows = 1536 floats = 6 KB
#define TILE_VEC4    (TILE_FLOATS / 4)   // 384 x 16B async transfers
#define PATCH_ELEMS  (PSIZE * PSIZE)     // 36
#define VEC4_PER_WG  (NPOS * (PATCH_ELEMS / 4))  // 126 * 9 = 1134 float4 stores

typedef float __attribute__((ext_vector_type(4))) v4f;

__global__ __launch_bounds__(256) void im2patch_kernel(const float* __restrict__ x,
                                                       float* __restrict__ out) {
    __shared__ float tile[TILE_FLOATS];

    const int p   = blockIdx.x;   // patch-row index, 0..125  (image row base = 2*p)
    const int b   = blockIdx.y;   // batch, 0..127
    const int tid = threadIdx.x;  // 0..255 (8 waves of 32)

    // Rows 2p .. 2p+5 of image b form ONE contiguous 6 KB block in memory.
    const float* src = x + (size_t)b * (IMG * IMG) + (size_t)(2 * p) * IMG;

    // ---- Stage 1: async DMA global -> LDS (CDNA5 GLOBAL_LOAD_ASYNC_TO_LDS_B128) ----
    // 384 x 16B transfers over 256 lanes; loop bound is wave-uniform so EXEC is
    // all-ones on every issued async op.
    for (int k = tid; k < TILE_VEC4; k += 256) {
        uint64_t gaddr = (uint64_t)(uintptr_t)(src + 4 * k);
        uint32_t laddr = (uint32_t)(uintptr_t)(&tile[4 * k]); // low 32 bits = LDS offset
        asm volatile("global_load_async_to_lds_b128 %0, %1, off"
                     :
                     : "v"(laddr), "v"(gaddr)
                     : "memory");
    }
    // Drain this wave's ASYNCcnt, then workgroup barrier so every wave sees LDS.
    asm volatile("s_wait_asynccnt 0" ::: "memory");
    __syncthreads();

    // ---- Stage 2: gather patches from LDS, non-temporal 128-bit stores ----
    // Output tile for this WG: 126 patches x 36 floats, contiguous, 16B aligned.
    float* obase = out + ((size_t)b * (NPOS * NPOS) + (size_t)p * NPOS) * PATCH_ELEMS;

    for (int f = tid; f < VEC4_PER_WG; f += 256) {
        const int q  = f / 9;          // patch-col index -> image col base 2*q
        const int k4 = f % 9;          // which float4 of the 36-float patch
        const int c0 = 2 * q;
        const int e0 = 4 * k4;

        v4f v;
#pragma unroll
        for (int t = 0; t < 4; ++t) {
            const int e  = e0 + t;     // element within patch, 0..35
            const int ri = e / 6;      // row within patch
            const int ci = e % 6;      // col within patch
            v[t] = tile[ri * IMG + c0 + ci];
        }
        __builtin_nontemporal_store(v, (v4f*)(obase + (size_t)q * PATCH_ELEMS + e0));
    }
}

extern "C" void kernel_launch(void* const* d_in, const int* in_sizes, int n_in,
                              void* d_out, int out_size, void* d_ws, size_t ws_size,
                              hipStream_t stream) {
    const float* x  = (const float*)d_in[0]; // (128,1,256,256) fp32
    float* out      = (float*)d_out;         // (128, 15876, 36) fp32
    (void)in_sizes; (void)n_in; (void)out_size; (void)d_ws; (void)ws_size;

    dim3 grid(NPOS, 128);   // (patch-row, batch)
    dim3 block(256);
    im2patch_kernel<<<grid, block, 0, stream>>>(x, out);
}